// PatchEmbed_60481729463054
// MI455X (gfx1250) — compile-verified
//
#include <hip/hip_runtime.h>

typedef __attribute__((ext_vector_type(2))) float v2f;
typedef __attribute__((ext_vector_type(8))) float v8f;

#define IMG   128
#define RC    32
#define ACNT  128
#define NRAD  5
#define NAZ   5
#define CIN   3
#define EMB   96
#define KDIM  75              // CIN * NRAD * NAZ
#define KPAD  76              // one zero pad column; LDS row stride (bank-conflict free)
#define MBLK  128             // points per block
#define NPTS  (RC * ACNT)     // 4096

__global__ __launch_bounds__(256)
void patch_embed_fused(const float* __restrict__ x,
                       const float* __restrict__ dist,
                       const float* __restrict__ Wg,
                       const float* __restrict__ bias,
                       float* __restrict__ out)
{
    // Aliased LDS: phase 1/2 use it as the feature tile [128 x 76] (38,912 B),
    // the epilogue reuses it as the output tile [128 x 96] (49,152 B).
    __shared__ __align__(16) float smem[MBLK * EMB];   // 49,152 B
    __shared__ float rlist[RC + 1];
    float* sfeat = smem;

    const int tid  = threadIdx.x;
    const int bimg = blockIdx.y;
    const int tile = blockIdx.x;

    const float xi  = dist[bimg * 3 + 0];
    const float fl  = dist[bimg * 3 + 1];
    const float fov = dist[2];             // reference uses dist[0, 2]

    // r_list[k] = f * sin(a)/(xi + cos(a)), a = atan(t_k)  ==  f*t/(1 + xi*sqrt(1+t^2))
    if (tid <= RC) {
        float t = ((float)tid * (1.0f / (float)RC)) * tanf(0.5f * fov);
        rlist[tid] = fl * t / (1.0f + xi * sqrtf(1.0f + t * t));
    }
    if (tid < MBLK) sfeat[tid * KPAD + KDIM] = 0.0f;   // zero the K-pad column
    __syncthreads();

    const float* img   = x + (size_t)bimg * (CIN * IMG * IMG);
    const int    pBase = tile * MBLK;
    const float  alpha = 6.283185307179586f / (float)ACNT;

    // ---- Phase 1: sample 128 points x 25 taps x 3 channels into LDS ----
    for (int t = tid; t < MBLK * NRAD * NAZ; t += 256) {
        int row = t / (NRAD * NAZ);
        int s   = t - row * (NRAD * NAZ);       // s = ri*NAZ + j
        int p   = pBase + row;
        int rc  = p & (RC - 1);
        int ac  = p >> 5;
        int ri  = s / NAZ;
        int j   = s - ri * NAZ;

        float r0 = rlist[rc], r1 = rlist[rc + 1];
        float radius = r0 + (r1 - r0) * (1.0f - 0.25f * (float)ri);
        float az = ((float)ac + 0.25f * (float)j) * alpha;
        float sz, cz;
        sincosf(az, &sz, &cz);

        // grid in [-1,1], then align_corners=True mapping to pixels
        float gx = (radius * cz * (1.0f / 64.0f) + 1.0f) * 63.5f;
        float gy = (radius * sz * (1.0f / 64.0f) + 1.0f) * 63.5f;

        float fx0 = floorf(gx), fy0 = floorf(gy);
        int   x0 = (int)fx0, y0 = (int)fy0;
        int   x1 = x0 + 1,   y1 = y0 + 1;
        float wx = gx - fx0, wy = gy - fy0;

        float vx0 = (x0 >= 0 && x0 < IMG) ? 1.0f : 0.0f;
        float vx1 = (x1 >= 0 && x1 < IMG) ? 1.0f : 0.0f;
        float vy0 = (y0 >= 0 && y0 < IMG) ? 1.0f : 0.0f;
        float vy1 = (y1 >= 0 && y1 < IMG) ? 1.0f : 0.0f;

        float w00 = (1.0f - wx) * (1.0f - wy) * vx0 * vy0;
        float w01 = wx          * (1.0f - wy) * vx1 * vy0;
        float w10 = (1.0f - wx) * wy          * vx0 * vy1;
        float w11 = wx          * wy          * vx1 * vy1;

        int cx0 = min(max(x0, 0), IMG - 1), cx1 = min(max(x1, 0), IMG - 1);
        int cy0 = min(max(y0, 0), IMG - 1), cy1 = min(max(y1, 0), IMG - 1);
        int i00 = cy0 * IMG + cx0, i01 = cy0 * IMG + cx1;
        int i10 = cy1 * IMG + cx0, i11 = cy1 * IMG + cx1;

        #pragma unroll
        for (int c = 0; c < CIN; ++c) {
            const float* ic = img + c * (IMG * IMG);
            float v = ic[i00] * w00 + ic[i01] * w01 + ic[i10] * w10 + ic[i11] * w11;
            sfeat[row * KPAD + c * (NRAD * NAZ) + s] = v;
        }
    }
    __syncthreads();

    // ---- Phase 2: per-wave 16x96 GEMM tile via V_WMMA_F32_16X16X4_F32 ----
    const int lane = tid & 31;
    const int wv   = tid >> 5;    // wave 0..7 -> 16-row strip
    const int lm   = lane & 15;   // N index (B/C/D), M index (A)
    const int lh   = lane >> 4;   // K-half select

    v8f acc[6];
    #pragma unroll
    for (int n = 0; n < 6; ++n) acc[n] = (v8f)0.0f;

    const float* arow = &sfeat[(wv * 16 + lm) * KPAD];

    // Clean steps: K rows 0..71, no pad masking, affine W addressing.
    for (int k = 0; k < 18; ++k) {
        int ka = k * 4 + lh * 2;            // A/B element K index for v=0
        v2f afrag;
        afrag.x = arow[ka];                 // ds_read_b64 (8B aligned)
        afrag.y = arow[ka + 1];

        const float* wrow = Wg + ka * EMB + lm;
        #pragma unroll
        for (int n = 0; n < 6; ++n) {
            v2f bfrag;
            bfrag.x = wrow[n * 16];         // W[ka,   col]
            bfrag.y = wrow[EMB + n * 16];   // W[ka+1, col]
            acc[n] = __builtin_amdgcn_wmma_f32_16x16x4_f32(
                         false, afrag, false, bfrag,
                         (short)0, acc[n], false, false);
        }
    }
    // Tail step: K rows 72..75 (row 75 is pad -> masked B, LDS col 75 is zero).
    {
        int ka = 72 + lh * 2;               // 72/73 (lh=0) or 74/75 (lh=1)
        v2f afrag;
        afrag.x = arow[ka];
        afrag.y = arow[ka + 1];

        int   kr1 = min(ka + 1, KDIM - 1);
        float m1  = (ka + 1 < KDIM) ? 1.0f : 0.0f;
        const float* wrow0 = Wg + ka  * EMB + lm;
        const float* wrow1 = Wg + kr1 * EMB + lm;
        #pragma unroll
        for (int n = 0; n < 6; ++n) {
            v2f bfrag;
            bfrag.x = wrow0[n * 16];
            bfrag.y = wrow1[n * 16] * m1;
            acc[n] = __builtin_amdgcn_wmma_f32_16x16x4_f32(
                         false, afrag, false, bfrag,
                         (short)0, acc[n], false, false);
        }
    }

    // ---- Epilogue: bias + stage tile in LDS, then coalesced b128 stores ----
    __syncthreads();   // all waves done reading sfeat; safe to alias as out tile
    #pragma unroll
    for (int n = 0; n < 6; ++n) {
        int   col = n * 16 + lm;
        float bn  = bias[col];
        #pragma unroll
        for (int i = 0; i < 8; ++i) {
            int m = wv * 16 + i + lh * 8;         // row within the 128-row tile
            smem[m * EMB + col] = acc[n][i] + bn;
        }
    }
    __syncthreads();

    const size_t outBase = ((size_t)bimg * NPTS + (size_t)pBase) * EMB;
    const float4* s4 = (const float4*)smem;
    float4*       o4 = (float4*)(out + outBase);
    #pragma unroll
    for (int it = 0; it < (MBLK * EMB) / (4 * 256); ++it) {   // 12 iters
        int idx = it * 256 + tid;
        o4[idx] = s4[idx];
    }
}

extern "C" void kernel_launch(void* const* d_in, const int* in_sizes, int n_in,
                              void* d_out, int out_size, void* d_ws, size_t ws_size,
                              hipStream_t stream) {
    const float* x    = (const float*)d_in[0];   // [B,3,128,128] f32
    const float* dist = (const float*)d_in[1];   // [B,3] f32
    const float* Wg   = (const float*)d_in[2];   // [75,96] f32
    const float* bias = (const float*)d_in[3];   // [96] f32
    float*       out  = (float*)d_out;           // [B,4096,96] f32

    int Bn = in_sizes[1] / 3;                    // 128
    dim3 grid(NPTS / MBLK, Bn);                  // (32, 128)
    patch_embed_fused<<<grid, 256, 0, stream>>>(x, dist, Wg, bias, out);
    (void)n_in; (void)out_size; (void)d_ws; (void)ws_size;
}